// EBSTM_34084860461440
// MI455X (gfx1250) — compile-verified
//
#include <hip/hip_runtime.h>
#include <hip/hip_bf16.h>
#include <math.h>

// Problem constants (B=16, C=64, K=128, F=6)
#define BN 16
#define CN 64
#define KN 128
#define FN 6
#define HORIZON_S 3.0f
#define EPSF 1e-6f

typedef float v2f __attribute__((ext_vector_type(2)));
typedef float v8f __attribute__((ext_vector_type(8)));

// ---------------------------------------------------------------------------
// Kernel 1: per-(b,c) graph work — rel codes, edit token, Kahn acyclicity.
// One 128-thread block (4 waves) per (b,c); adjacency as row+col bitmasks in
// LDS, built with wave32 ballots (gfx1250-native) instead of per-bit atomics.
// ---------------------------------------------------------------------------
__global__ __launch_bounds__(KN) void graph_kernel(
    const float* __restrict__ A_t,      // (B,K,K)
    const float* __restrict__ cand,     // (B,C,K,K)
    const float* __restrict__ amask,    // (B,K)
    float* __restrict__ out_tokens,     // (B,C,4)
    float* __restrict__ out_rel,        // (B,C,K,K)
    float* __restrict__ out_acyc)       // (B,C)
{
    const int bc   = blockIdx.x;
    const int b    = bc >> 6;            // / CN
    const int c    = bc & (CN - 1);
    const int tid  = threadIdx.x;        // 0..127 ; acts as column index j
    const int wid  = tid >> 5;           // wave id: which 32-wide j-word
    const int lane = tid & 31;

    __shared__ unsigned rowC[KN][4];     // candidate rows  (bit j of row i)
    __shared__ unsigned colC[KN][4];     // candidate cols  (bit i of col j)
    __shared__ unsigned rowB[KN][4];     // base rows
    __shared__ unsigned colB[KN][4];     // base cols
    __shared__ unsigned remW[4];         // Kahn frontier bitmask
    __shared__ unsigned remFinal[4], valFinal[4];
    __shared__ int sMin;

    if (tid == 0) sMin = 0x7fffffff;

    const float* candbc = cand + (((size_t)b * CN + c) * KN) * KN;
    const float* basep  = A_t  + ((size_t)b * KN) * KN;

    // Binarize in one streaming pass. Thread tid owns column j=tid in
    // registers; each wave covers 32 consecutive j, so a single wave32
    // ballot IS the 32-bit row-word — one ds_store per wave per row.
    unsigned cw[4] = {0u, 0u, 0u, 0u};   // candidate column j bits (index i)
    unsigned bw[4] = {0u, 0u, 0u, 0u};   // base column j bits
    for (int i = 0; i < KN; ++i) {
        if (i + 8 < KN)
            __builtin_prefetch(candbc + (size_t)(i + 8) * KN + tid, 0, 1);
        const bool cbit = candbc[(size_t)i * KN + tid] > 0.5f;
        const bool bbit = basep [(size_t)i * KN + tid] > 0.5f;
        const unsigned cbal = __builtin_amdgcn_ballot_w32(cbit);
        const unsigned bbal = __builtin_amdgcn_ballot_w32(bbit);
        if (lane == 0) { rowC[i][wid] = cbal; rowB[i][wid] = bbal; }
        if (cbit) cw[i >> 5] |= 1u << (i & 31);
        if (bbit) bw[i >> 5] |= 1u << (i & 31);
    }
    for (int w = 0; w < 4; ++w) { colC[tid][w] = cw[w]; colB[tid][w] = bw[w]; }
    __syncthreads();

    // ---- rel = cb*1 + cbT*2 + (cb&cbT)*3  ==  cb + 2*cbT  (coalesced over j)
    {
        float* relp = out_rel + (((size_t)b * CN + c) * KN) * KN;
        const int wsel = tid >> 5, bsel = tid & 31;
        for (int i = 0; i < KN; ++i) {
            unsigned x = (rowC[i][wsel] >> bsel) & 1u;     // cb[i][j]
            unsigned y = (colC[i][wsel] >> bsel) & 1u;     // cb[j][i]
            relp[(size_t)i * KN + tid] = (float)(x + 2u * y);
        }
    }

    // ---- edit token: first (row-major, j>i) pair where diff | diffT set
    {
        const int i = tid;
        int best = 0x7fffffff;
        for (int w = 0; w < 4; ++w) {
            unsigned pd = (rowC[i][w] ^ rowB[i][w]) | (colC[i][w] ^ colB[i][w]);
            const int lo = w << 5;
            unsigned m;
            if (i < lo)            m = 0xffffffffu;        // whole word is j>i
            else if (i - lo >= 31) m = 0u;                 // no j>i in word
            else                   m = ~((2u << (i - lo)) - 1u);
            pd &= m;
            if (pd) { best = i * KN + (lo + __builtin_ctz(pd)); break; }
        }
        atomicMin(&sMin, best);
    }
    __syncthreads();
    if (tid == 0) {
        float t0 = 0.f, t1 = 0.f, t2 = 0.f, t3 = 0.f;
        if (sMin != 0x7fffffff) {
            int fi = sMin >> 7, fj = sMin & (KN - 1);
            int wsel = fj >> 5, bsel = fj & 31;
            int b_ij = (rowB[fi][wsel] >> bsel) & 1;
            int b_ji = (colB[fi][wsel] >> bsel) & 1;       // base[fj][fi]
            int c_ij = (rowC[fi][wsel] >> bsel) & 1;
            int c_ji = (colC[fi][wsel] >> bsel) & 1;
            int prev = (b_ij && !b_ji) ? 1 : ((b_ji && !b_ij) ? 2 : 0);
            int nxt  = (c_ij && !c_ji) ? 1 : ((c_ji && !c_ij) ? 2 : 0);
            t0 = (float)fi; t1 = (float)fj; t2 = (float)prev; t3 = (float)nxt;
        }
        float* tk = out_tokens + (size_t)bc * 4;
        tk[0] = t0; tk[1] = t1; tk[2] = t2; tk[3] = t3;
    }

    // ---- Kahn peeling: thread tid = node j; column bits live in cw[].
    // indeg_j = (sum over ALL i of A[i][j]) * valid_j  (reference semantics)
    const bool validj = amask[(size_t)b * KN + tid] > 0.5f;
    int indeg = validj
        ? (__popc(cw[0]) + __popc(cw[1]) + __popc(cw[2]) + __popc(cw[3])) : 0;
    bool removed = false;
    for (int t = 0; t < KN; ++t) {
        const bool rem = validj && !removed && (indeg == 0);
        const unsigned bal = __builtin_amdgcn_ballot_w32(rem);
        if (lane == 0) remW[wid] = bal;
        __syncthreads();
        const unsigned r0 = remW[0], r1 = remW[1], r2 = remW[2], r3 = remW[3];
        if ((r0 | r1 | r2 | r3) == 0u) break;   // uniform: fixed point
        if (rem) removed = true;
        indeg -= __popc(r0 & cw[0]) + __popc(r1 & cw[1])
               + __popc(r2 & cw[2]) + __popc(r3 & cw[3]);
        __syncthreads();                         // remW reads vs next write
    }
    // counts via ballots (no atomics)
    {
        const unsigned rb = __builtin_amdgcn_ballot_w32(removed);
        const unsigned vb = __builtin_amdgcn_ballot_w32(validj);
        if (lane == 0) { remFinal[wid] = rb; valFinal[wid] = vb; }
    }
    __syncthreads();
    if (tid == 0) {
        int remCnt = __popc(remFinal[0]) + __popc(remFinal[1])
                   + __popc(remFinal[2]) + __popc(remFinal[3]);
        int vCnt   = __popc(valFinal[0]) + __popc(valFinal[1])
                   + __popc(valFinal[2]) + __popc(valFinal[3]);
        out_acyc[bc] = (float)((remCnt == vCnt) || (vCnt <= 1));
    }
}

// ---------------------------------------------------------------------------
// Kernel 2: pairwise dists via V_WMMA_F32_16X16X4_F32 Gram matrices.
// One 256-thread block per b; each wave computes 8 of the 64 16x16 tiles.
// d^2  = pp[i]+pp[j]-2 P.P^T[i,j]
// vv   = vv[i]+vv[j]-2 V.V^T[i,j]
// rv   = pv[i]+pv[j]-  P.V^T[i,j] - V.P^T[i,j]
// Fast-math epilogue: v_rcp_f32 / v_sqrt_f32 (TRANS ops) instead of the
// correctly-rounded div/sqrt expansions — ~1 op vs ~12 per element.
// ---------------------------------------------------------------------------
__global__ __launch_bounds__(256) void dists_kernel(
    const float* __restrict__ feat,    // (B,K,F)
    float* __restrict__ out_d,         // (B,K,K)
    float* __restrict__ out_dmin)      // (B,K,K)
{
    const int b   = blockIdx.x;
    const int tid = threadIdx.x;
    __shared__ float px[KN], py[KN], vx[KN], vy[KN], pp[KN], vv[KN], pv[KN];

    if (tid < KN) {
        const float* f = feat + ((size_t)b * KN + tid) * FN;
        float ax = f[0], ay = f[1], bx = f[3], by = f[4];
        px[tid] = ax; py[tid] = ay; vx[tid] = bx; vy[tid] = by;
        pp[tid] = ax * ax + ay * ay;
        vv[tid] = bx * bx + by * by;
        pv[tid] = ax * bx + ay * by;
    }
    __syncthreads();

    const int wave = tid >> 5;
    const int lane = tid & 31;
    for (int t = wave; t < 64; t += 8) {     // uniform per wave: 8 tiles each
        const int m0 = (t >> 3) << 4;        // tile row origin (i)
        const int n0 = (t & 7) << 4;         // tile col origin (j)

        // A 16x4 f32 operand: lanes 0-15 -> M=lane, K=0/1; lanes 16-31 -> K=2/3 pad=0
        v2f aP = {0.f, 0.f}, aV = {0.f, 0.f}, bP = {0.f, 0.f}, bV = {0.f, 0.f};
        if (lane < 16) {
            aP.x = px[m0 + lane]; aP.y = py[m0 + lane];
            aV.x = vx[m0 + lane]; aV.y = vy[m0 + lane];
            bP.x = px[n0 + lane]; bP.y = py[n0 + lane];
            bV.x = vx[n0 + lane]; bV.y = vy[n0 + lane];
        }
        v8f z = {};
        v8f gpp = __builtin_amdgcn_wmma_f32_16x16x4_f32(false, aP, false, bP, (short)0, z, false, false);
        v8f gvv = __builtin_amdgcn_wmma_f32_16x16x4_f32(false, aV, false, bV, (short)0, z, false, false);
        v8f gpv = __builtin_amdgcn_wmma_f32_16x16x4_f32(false, aP, false, bV, (short)0, z, false, false);
        v8f gvp = __builtin_amdgcn_wmma_f32_16x16x4_f32(false, aV, false, bP, (short)0, z, false, false);

        // C/D layout: VGPR r, lane<16: (M=r, N=lane) ; lane>=16: (M=8+r, N=lane-16)
        const int N     = (lane < 16) ? lane : (lane - 16);
        const int Mbase = (lane < 16) ? 0 : 8;
        for (int r = 0; r < 8; ++r) {
            const int i = m0 + Mbase + r;
            const int j = n0 + N;
            float dd  = fmaxf(pp[i] + pp[j] - 2.0f * gpp[r], 0.0f);
            float vvs = vv[i] + vv[j] - 2.0f * gvv[r];
            float rv  = pv[i] + pv[j] - gpv[r] - gvp[r];
            float ts  = -rv * __builtin_amdgcn_rcpf(vvs + EPSF);   // v_rcp_f32
            ts = fminf(fmaxf(ts, 0.0f), HORIZON_S);
            float dm2 = fmaxf(dd + 2.0f * rv * ts + vvs * ts * ts, 0.0f);
            float dmin = (i == j) ? __builtin_inff()
                                  : __builtin_amdgcn_sqrtf(dm2);   // v_sqrt_f32
            const size_t off = ((size_t)b * KN + i) * KN + j;
            out_d[off]    = __builtin_amdgcn_sqrtf(dd);
            out_dmin[off] = dmin;
        }
    }
}

// ---------------------------------------------------------------------------
extern "C" void kernel_launch(void* const* d_in, const int* in_sizes, int n_in,
                              void* d_out, int out_size, void* d_ws, size_t ws_size,
                              hipStream_t stream) {
    (void)in_sizes; (void)n_in; (void)d_ws; (void)ws_size; (void)out_size;

    const float* A_t   = (const float*)d_in[0];   // (B,K,K)
    const float* cand  = (const float*)d_in[1];   // (B,C,K,K)
    const float* feat  = (const float*)d_in[2];   // (B,K,F)
    const float* amask = (const float*)d_in[3];   // (B,K)

    float* out = (float*)d_out;
    const size_t TOK  = (size_t)BN * CN * 4;             // 4096
    const size_t REL  = (size_t)BN * CN * KN * KN;       // 16777216
    const size_t ACY  = (size_t)BN * CN;                 // 1024
    const size_t DK   = (size_t)BN * KN * KN;            // 262144
    float* out_tokens = out;
    float* out_rel    = out + TOK;
    float* out_acyc   = out + TOK + REL;
    float* out_d      = out + TOK + REL + ACY;
    float* out_dmin   = out_d + DK;

    graph_kernel<<<dim3(BN * CN), dim3(KN), 0, stream>>>(
        A_t, cand, amask, out_tokens, out_rel, out_acyc);
    dists_kernel<<<dim3(BN), dim3(256), 0, stream>>>(feat, out_d, out_dmin);
}